// DistanceNMS_81801947119862
// MI455X (gfx1250) — compile-verified
//
#include <hip/hip_runtime.h>

typedef __attribute__((ext_vector_type(2))) float v2f;
typedef __attribute__((ext_vector_type(8))) float v8f;

constexpr int   NPTS  = 4096;          // peaks per batch
constexpr int   NT    = 256;           // threads per block (8 waves of 32)
constexpr int   NTILE = NPTS / 16;     // 256 row/col tiles
constexpr float D2T   = 16.0f;         // NMS_DIST^2 = 4.0^2
constexpr float FAR   = 1.0e15f;       // sentinel for suppressed pivot rows

__device__ __forceinline__ float min8(const v8f& d) {
    float a = fminf(d[0], d[1]);
    float b = fminf(d[2], d[3]);
    float c = fminf(d[4], d[5]);
    float e = fminf(d[6], d[7]);
    return fminf(fminf(a, b), fminf(c, e));
}

__global__ __launch_bounds__(256)
void DistanceNMS_kernel(const float* __restrict__ peaks, float* __restrict__ out) {
    __shared__ float sx[NPTS];
    __shared__ float sy[NPTS];
    __shared__ float sc[NPTS];
    __shared__ unsigned char keep[NPTS];

    const int tid   = threadIdx.x;
    const int batch = blockIdx.x;
    const float* p  = peaks + (size_t)batch * NPTS * 3;

    // ---- load peaks into LDS (SoA) ----
    for (int i = tid; i < NPTS; i += NT) {
        sx[i] = p[i * 3 + 0];
        sy[i] = p[i * 3 + 1];
        sc[i] = p[i * 3 + 2];
    }
    __syncthreads();

    // ---- bitonic sort by confidence, descending ----
    for (int k = 2; k <= NPTS; k <<= 1) {
        for (int j = k >> 1; j > 0; j >>= 1) {
            for (int i = tid; i < NPTS; i += NT) {
                int ixj = i ^ j;
                if (ixj > i) {
                    bool up = ((i & k) == 0);
                    float ci = sc[i], cj = sc[ixj];
                    bool sw = up ? (ci < cj) : (ci > cj);   // descending final order
                    if (sw) {
                        float xi = sx[i], yi = sy[i];
                        sx[i] = sx[ixj]; sy[i] = sy[ixj]; sc[i] = cj;
                        sx[ixj] = xi;    sy[ixj] = yi;    sc[ixj] = ci;
                    }
                }
            }
            __syncthreads();
        }
    }

    // ---- init keep flags ----
    for (int i = tid; i < NPTS; i += NT) keep[i] = 1;
    __syncthreads();

    const int  wave = tid >> 5;
    const int  lane = tid & 31;
    const int  l15  = lane & 15;
    const bool hi   = lane >= 16;

    // ---- blocked greedy NMS: 256 serial pivot steps ----
    for (int b = 0; b < NTILE; ++b) {
        const int base = b * 16;

        // (1) resolve the 16x16 diagonal block sequentially (tiny)
        if (tid == 0) {
            float px[16], py[16];
            unsigned char kp[16];
            for (int i = 0; i < 16; ++i) {
                px[i] = sx[base + i];
                py[i] = sy[base + i];
                kp[i] = keep[base + i];
            }
            for (int i = 0; i < 16; ++i) {
                if (!kp[i]) continue;
                for (int j = i + 1; j < 16; ++j) {
                    float dx = px[j] - px[i], dy = py[j] - py[i];
                    if (dx * dx + dy * dy < D2T) kp[j] = 0;
                }
            }
            for (int i = 0; i < 16; ++i) keep[base + i] = kp[i];
        }
        __syncthreads();

        // (2) finalized pivot rows suppress all later columns via WMMA d2 tiles.
        // A row m = (x_m, y_m, |p_m|^2, 1); B col n = (-2x_n, -2y_n, 1, |p_n|^2)
        // => D[m][n] = |p_m - p_n|^2 exactly, one v_wmma_f32_16x16x4_f32 per tile.
        // Dead pivot rows are moved to a far sentinel so the suppression test is
        // simply min(D row block) < thresh -- no per-row masking needed.
        const int mrow = base + l15;
        float mx = sx[mrow], my = sy[mrow];
        if (!keep[mrow]) { mx = FAR; my = 0.0f; }
        v2f A;
        if (!hi) { A.x = mx;                A.y = my;   }   // lanes 0-15: K0,K1
        else     { A.x = mx * mx + my * my; A.y = 1.0f; }   // lanes 16-31: K2,K3

        int ct = b + 1 + wave;                 // wave-uniform bounds
        // 2x unrolled: two independent WMMA tiles in flight to hide D latency
        for (; ct + 8 < NTILE; ct += 16) {
            const int n0 = ct * 16 + l15;
            const int n1 = (ct + 8) * 16 + l15;
            const float x0 = sx[n0], y0 = sy[n0];
            const float x1 = sx[n1], y1 = sy[n1];
            v2f B0, B1;
            if (!hi) { B0.x = -2.0f * x0; B0.y = -2.0f * y0;
                       B1.x = -2.0f * x1; B1.y = -2.0f * y1; }
            else     { B0.x = 1.0f;       B0.y = x0 * x0 + y0 * y0;
                       B1.x = 1.0f;       B1.y = x1 * x1 + y1 * y1; }
            v8f C0 = {}, C1 = {};
            v8f D0 = __builtin_amdgcn_wmma_f32_16x16x4_f32(
                false, A, false, B0, (short)0, C0, false, false);
            v8f D1 = __builtin_amdgcn_wmma_f32_16x16x4_f32(
                false, A, false, B1, (short)0, C1, false, false);
            if (min8(D0) < D2T) keep[n0] = 0;   // races only write 0 -> benign
            if (min8(D1) < D2T) keep[n1] = 0;
        }
        if (ct < NTILE) {
            const int n0 = ct * 16 + l15;
            const float x0 = sx[n0], y0 = sy[n0];
            v2f B0;
            if (!hi) { B0.x = -2.0f * x0; B0.y = -2.0f * y0; }
            else     { B0.x = 1.0f;       B0.y = x0 * x0 + y0 * y0; }
            v8f C0 = {};
            v8f D0 = __builtin_amdgcn_wmma_f32_16x16x4_f32(
                false, A, false, B0, (short)0, C0, false, false);
            if (min8(D0) < D2T) keep[n0] = 0;
        }
        __syncthreads();
    }

    // ---- emit: sorted peaks masked by keep ----
    for (int i = tid; i < NPTS; i += NT) {
        float m = keep[i] ? 1.0f : 0.0f;
        float* o = out + (size_t)batch * NPTS * 3 + (size_t)i * 3;
        o[0] = sx[i] * m;
        o[1] = sy[i] * m;
        o[2] = sc[i] * m;
    }
}

extern "C" void kernel_launch(void* const* d_in, const int* in_sizes, int n_in,
                              void* d_out, int out_size, void* d_ws, size_t ws_size,
                              hipStream_t stream) {
    (void)n_in; (void)out_size; (void)d_ws; (void)ws_size;
    const float* peaks = (const float*)d_in[0];
    float* out = (float*)d_out;
    const int nbatch = in_sizes[0] / (NPTS * 3);   // 16 for the reference shapes
    DistanceNMS_kernel<<<dim3(nbatch), dim3(NT), 0, stream>>>(peaks, out);
}